// KmeansNN_11665131176009
// MI455X (gfx1250) — compile-verified
//
#include <hip/hip_runtime.h>

// Problem constants (match reference)
#define MSTAGES 8
#define KCODES  256
#define DDIM    128
#define BATCH   65536
#define BT      32      // batch rows per block
#define NWAVES  8       // 256 threads / wave32

typedef float v2f __attribute__((ext_vector_type(2)));
typedef float v8f __attribute__((ext_vector_type(8)));

// Workspace layout (floats)
// cenPT: GEMM1 B operand, pair-interleaved transpose: cenPT[p][n][2] = cen[n][2p+i]
// cenPJ: GEMM2 B operand, pair-interleaved:           cenPJ[p][d][2] = cen[2p+i][d]
#define WS_CENPT 0
#define WS_CENPJ (WS_CENPT + DDIM * KCODES)
#define WS_CEN2  (WS_CENPJ + KCODES * DDIM)
#define WS_LOSS  (WS_CEN2 + KCODES)            // [M][3] : hard, soft, match

static __device__ __forceinline__ v8f wmma_f32_k4(v2f a, v2f b, v8f c) {
  // V_WMMA_F32_16X16X4_F32 : D = A(16x4,f32) * B(4x16,f32) + C(16x16,f32)
  return __builtin_amdgcn_wmma_f32_16x16x4_f32(
      false, a, false, b, (short)0, c, false, false);
}

// Per-stage prep: packed B layouts; cen2[k] = sum_d c^2; zero loss slots.
__global__ void prep_stage(const float* __restrict__ cenJ,
                           float* __restrict__ ws, int j) {
  const int k = blockIdx.x;     // 0..255 (code)
  const int d = threadIdx.x;    // 0..127 (dim)
  float c = cenJ[k * DDIM + d];
  // GEMM1 B: lane needs (cenT[2p][n], cenT[2p+1][n]) contiguous
  ws[WS_CENPT + (d >> 1) * (2 * KCODES) + k * 2 + (d & 1)] = c;
  // GEMM2 B: lane needs (cen[2p][d], cen[2p+1][d]) contiguous
  ws[WS_CENPJ + (k >> 1) * (2 * DDIM) + d * 2 + (k & 1)] = c;
  float s = c * c;
  #pragma unroll
  for (int o = 16; o > 0; o >>= 1) s += __shfl_xor(s, o, 32);
  __shared__ float part[4];
  const int wv = d >> 5, ln = d & 31;
  if (ln == 0) part[wv] = s;
  __syncthreads();
  if (d == 0) ws[WS_CEN2 + k] = part[0] + part[1] + part[2] + part[3];
  if (blockIdx.x == 0 && d < 3) ws[WS_LOSS + j * 3 + d] = 0.0f;
}

__global__ __launch_bounds__(256) void rvq_stage(
    const float* __restrict__ x,
    const float* __restrict__ cenJ,      // center[j] : [K][D]
    float* __restrict__ ws,
    float* __restrict__ xr_out,          // X_r_matrix [B][M][D]
    float* __restrict__ xp_out,          // X_p_matrix (residuals) [B][M][D]
    float* __restrict__ xpm_out,         // X_p_m accumulator / output [B][D]
    float* __restrict__ xpm2_out,        // duplicate X_p_m output
    int*   __restrict__ codes_out,       // [B][M] int32
    int j, int is_last)
{
  __shared__ float sRX[BT][132];     // residual tile, padded (conflict-free)
  __shared__ float sLog[BT][260];    // logits -> soft probs, padded
  __shared__ float sRx2[BT];
  __shared__ float sCen2[KCODES];
  __shared__ int   sIdx[BT];

  const int t    = threadIdx.x;
  const int wv   = t >> 5;
  const int ln   = t & 31;
  const int half = ln >> 4;          // WMMA lane split (K pairs)
  const int m16  = ln & 15;
  const long b0  = (long)blockIdx.x * BT;
  const float* __restrict__ cenPT = ws + WS_CENPT;
  const float* __restrict__ cenPJ = ws + WS_CENPJ;

  // ---- Phase 0: residual RX = x - X_p_m ; write residual output (coalesced)
  sCen2[t] = ws[WS_CEN2 + t];
  #pragma unroll
  for (int i = 0; i < (BT * DDIM) / 256; ++i) {
    int e  = t + i * 256;
    int bl = e >> 7;                 // /128
    int d  = e & 127;
    long g = (b0 + bl) * DDIM + d;
    float rx = x[g];
    if (j > 0) rx -= xpm_out[g];
    sRX[bl][d] = rx;
    xp_out[((b0 + bl) * MSTAGES + j) * (long)DDIM + d] = rx;
  }
  __syncthreads();
  // row ||rx||^2 : each wave handles 4 rows
  #pragma unroll
  for (int rr = 0; rr < 4; ++rr) {
    int r = wv * 4 + rr;
    float s = 0.0f;
    #pragma unroll
    for (int q = 0; q < DDIM / 32; ++q) {
      float v = sRX[r][ln + 32 * q];
      s += v * v;
    }
    #pragma unroll
    for (int o = 16; o > 0; o >>= 1) s += __shfl_xor(s, o, 32);
    if (ln == 0) sRx2[r] = s;
  }
  __syncthreads();

  // ---- Phase 1: GEMM1 (fp32 WMMA): logits[b,k] = -sqrt(max(d2,0))
  #pragma unroll
  for (int it = 0; it < 4; ++it) {
    int tile = wv + NWAVES * it;     // 32 tiles: 2 row-blocks x 16 code-blocks
    int r0 = (tile >> 4) * 16;
    int n0 = (tile & 15) * 16;
    v8f acc = {0.f, 0.f, 0.f, 0.f, 0.f, 0.f, 0.f, 0.f};
    #pragma unroll 4
    for (int k0 = 0; k0 < DDIM; k0 += 4) {
      int kb = k0 + half * 2;
      int p  = (k0 >> 1) + half;     // d-pair index
      v2f a; a.x = sRX[r0 + m16][kb]; a.y = sRX[r0 + m16][kb + 1];
      v2f b = *(const v2f*)(cenPT + p * (2 * KCODES) + (n0 + m16) * 2);
      acc = wmma_f32_k4(a, b, acc);
    }
    #pragma unroll
    for (int i = 0; i < 8; ++i) {
      int gr = r0 + i + half * 8;
      int gc = n0 + m16;
      float d2 = sRx2[gr] - 2.0f * acc[i] + sCen2[gc];
      d2 = fmaxf(d2, 0.0f);
      sLog[gr][gc] = -__builtin_amdgcn_sqrtf(d2);      // T = 1
    }
  }
  __syncthreads();

  // ---- Phase 2: softmax + argmax per row; l_match = sum p^2 - 2 p_idx + 1
  float lmatch_w = 0.0f;
  #pragma unroll
  for (int rr = 0; rr < 4; ++rr) {
    int r = wv * 4 + rr;
    float vals[KCODES / 32];
    float vmax = -3.4e38f; int vidx = KCODES;
    #pragma unroll
    for (int q = 0; q < KCODES / 32; ++q) {
      int k = ln + 32 * q;
      float v = sLog[r][k];
      vals[q] = v;
      if (v > vmax) { vmax = v; vidx = k; }        // first-max within lane
    }
    #pragma unroll
    for (int o = 16; o > 0; o >>= 1) {
      float ov = __shfl_xor(vmax, o, 32);
      int   oi = __shfl_xor(vidx, o, 32);
      if (ov > vmax || (ov == vmax && oi < vidx)) { vmax = ov; vidx = oi; }
    }
    float es = 0.0f;
    #pragma unroll
    for (int q = 0; q < KCODES / 32; ++q) { vals[q] = __expf(vals[q] - vmax); es += vals[q]; }
    #pragma unroll
    for (int o = 16; o > 0; o >>= 1) es += __shfl_xor(es, o, 32);
    float inv = 1.0f / es;
    float lm = 0.0f;
    #pragma unroll
    for (int q = 0; q < KCODES / 32; ++q) {
      int k = ln + 32 * q;
      float p = vals[q] * inv;
      sLog[r][k] = p;                              // overwrite logits with soft
      lm += p * p;
      if (k == vidx) lm += 1.0f - 2.0f * p;
    }
    #pragma unroll
    for (int o = 16; o > 0; o >>= 1) lm += __shfl_xor(lm, o, 32);
    if (ln == 0) {
      sIdx[r] = vidx;
      codes_out[(b0 + r) * MSTAGES + j] = vidx;
      lmatch_w += lm;
    }
  }
  if (ln == 0) atomicAdd(&ws[WS_LOSS + j * 3 + 2], lmatch_w);
  __syncthreads();

  // ---- Phase 3: GEMM2 soft@cen (WMMA), hard gather, losses, outputs
  float lsoft = 0.0f, lhard = 0.0f;
  #pragma unroll
  for (int it = 0; it < 2; ++it) {
    int tile = wv + NWAVES * it;     // 16 tiles: 2 row-blocks x 8 d-blocks
    int r0 = (tile >> 3) * 16;
    int n0 = (tile & 7) * 16;
    v8f acc = {0.f, 0.f, 0.f, 0.f, 0.f, 0.f, 0.f, 0.f};
    #pragma unroll 4
    for (int k0 = 0; k0 < KCODES; k0 += 4) {
      int kb = k0 + half * 2;
      int p  = (k0 >> 1) + half;     // k-pair index
      v2f a; a.x = sLog[r0 + m16][kb]; a.y = sLog[r0 + m16][kb + 1];
      v2f b = *(const v2f*)(cenPJ + p * (2 * DDIM) + (n0 + m16) * 2);
      acc = wmma_f32_k4(a, b, acc);
    }
    #pragma unroll
    for (int i = 0; i < 8; ++i) {
      int rl = r0 + i + half * 8;
      long b  = b0 + rl;
      int  d  = n0 + m16;
      float rxv = sRX[rl][d];
      float ds  = acc[i] - rxv;      lsoft += ds * ds;   // soft recon loss
      int   idx = sIdx[rl];
      float xpv = cenJ[idx * DDIM + d];                  // hard recon (coalesced)
      float dh  = xpv - rxv;         lhard += dh * dh;
      xr_out[(b * MSTAGES + j) * (long)DDIM + d] = xpv;
      float nx = x[b * DDIM + d] - rxv + xpv;            // new X_p_m
      xpm_out[b * DDIM + d] = nx;
      if (is_last) xpm2_out[b * DDIM + d] = nx;
    }
  }
  #pragma unroll
  for (int o = 16; o > 0; o >>= 1) {
    lsoft += __shfl_xor(lsoft, o, 32);
    lhard += __shfl_xor(lhard, o, 32);
  }
  if (ln == 0) {
    atomicAdd(&ws[WS_LOSS + j * 3 + 0], lhard);
    atomicAdd(&ws[WS_LOSS + j * 3 + 1], lsoft);
  }
}

__global__ void copy_codebooks(const float* __restrict__ center,
                               float* __restrict__ cb) {
  long i = (long)blockIdx.x * 256 + threadIdx.x;
  cb[i] = center[i];
}

__global__ void finalize(const float* __restrict__ ws,
                         const float* __restrict__ weight,
                         float* __restrict__ outp) {
  if (threadIdx.x == 0) {
    const float invBD = 1.0f / ((float)BATCH * (float)DDIM);
    const float invBK = 1.0f / ((float)BATCH * (float)KCODES);
    float acc = 0.0f;
    for (int jj = 0; jj < MSTAGES; ++jj) {
      float lh = ws[WS_LOSS + jj * 3 + 0] * invBD;
      float ls = ws[WS_LOSS + jj * 3 + 1] * invBD;
      float lm = ws[WS_LOSS + jj * 3 + 2] * invBK;
      acc += weight[jj] * (lh + ls + 0.1f * lm);
    }
    *outp = acc;
  }
}

extern "C" void kernel_launch(void* const* d_in, const int* in_sizes, int n_in,
                              void* d_out, int out_size, void* d_ws, size_t ws_size,
                              hipStream_t stream) {
  (void)in_sizes; (void)n_in; (void)out_size; (void)ws_size;
  const float* x      = (const float*)d_in[0];   // [B,D]
  const float* center = (const float*)d_in[1];   // [M,K,D]
  const float* weight = (const float*)d_in[2];   // [1,M]
  float* out = (float*)d_out;
  float* ws  = (float*)d_ws;

  const long NB_MD = (long)BATCH * MSTAGES * DDIM;   // 67,108,864
  const long NBD   = (long)BATCH * DDIM;             // 8,388,608
  float* xr    = out;                                // X_r_matrix
  float* xp    = out + NB_MD;                        // X_p_matrix
  float* xpm   = out + 2 * NB_MD;                    // X_p_m
  float* xpm2  = out + 2 * NB_MD + NBD;              // X_p_m (dup)
  float* cb    = out + 2 * NB_MD + 2 * NBD;          // codebooks
  int*   codes = (int*)(out + 2 * NB_MD + 2 * NBD + (long)MSTAGES * KCODES * DDIM);
  float* outp  = out + 2 * NB_MD + 2 * NBD + (long)MSTAGES * KCODES * DDIM
                     + (long)BATCH * MSTAGES;

  for (int j = 0; j < MSTAGES; ++j) {
    const float* cenJ = center + (long)j * KCODES * DDIM;
    prep_stage<<<KCODES, DDIM, 0, stream>>>(cenJ, ws, j);
    rvq_stage<<<BATCH / BT, 256, 0, stream>>>(x, cenJ, ws, xr, xp, xpm, xpm2,
                                              codes, j, j == MSTAGES - 1 ? 1 : 0);
  }
  copy_codebooks<<<(MSTAGES * KCODES * DDIM) / 256, 256, 0, stream>>>(center, cb);
  finalize<<<1, 32, 0, stream>>>(ws, weight, outp);
}